// SAGraphTransformerNet_23948737642620
// MI455X (gfx1250) — compile-verified
//
#include <hip/hip_runtime.h>
#include <hip/hip_bf16.h>
#include <math.h>

typedef float v2f __attribute__((ext_vector_type(2)));
typedef float v8f __attribute__((ext_vector_type(8)));

#define NN 50000
#define EE 800000
#define GG 128
#define DD 128
#define NHD 8
#define NL 4
#define MEIG 16
#define LPED 16
#define LPEFF 64
#define NCLS 10

// ----------------------------------------------------------------------
__global__ __launch_bounds__(256) void zero_f32(float* __restrict__ p, long n)
{
    long i = (long)blockIdx.x * blockDim.x + threadIdx.x;
    if (i < n) p[i] = 0.0f;
}

// ----------------------------------------------------------------------
// fp32 WMMA GEMM:  C[row, colOff+col] = act( A[row, 0:K] @ W[K, Nc] + bias ) (+resid)
// A lda == K. Block = 4 waves, 32-row slab; each wave computes a 32x32 output
// patch as 2x2 grid of 16x16 WMMA tiles (4 accumulators) so each A ds_load and
// each B global fetch feeds two v_wmma_f32_16x16x4_f32 ops.
// WMMA 16x16x4 f32 lane layouts per CDNA5 ISA §7.12.2:
//   A: lane<16 row=lane, {K+0,K+1}; lane>=16 row=lane-16, {K+2,K+3}
//   B: lane<16 col=lane, {K+0,K+1}; lane>=16 col=lane-16, {K+2,K+3}
//   C/D: VGPR i -> row i (lanes 0-15) / row 8+i (lanes 16-31), col = lane&15
__global__ __launch_bounds__(128) void gemm_wmma_f32(
    const float* __restrict__ A, const float* __restrict__ W,
    const float* __restrict__ bias, const float* __restrict__ resid,
    float* __restrict__ C, int M, int K, int Nc, int ldc, int colOff, int relu)
{
    __shared__ float sA[32 * 260];          // K<=256, row stride K+4 (bank-safe)
    const int stride = K + 4;
    const int rowBase = blockIdx.x << 5;
    const int lane = threadIdx.x & 31;
    const int wave = threadIdx.x >> 5;
    const int kq = K >> 2;

    // cooperative load of the 32 x K A-slab (float4, coalesced); clamp tail rows
    for (int idx = threadIdx.x; idx < 32 * kq; idx += blockDim.x) {
        int r = idx / kq, cq = idx - r * kq;
        int gr = rowBase + r; if (gr >= M) gr = M - 1;
        float4 v = ((const float4*)(A + (size_t)gr * K))[cq];
        *(float4*)&sA[r * stride + (cq << 2)] = v;
    }
    __syncthreads();

    const int nT = Nc >> 4;
    const int ct0 = ((blockIdx.y << 2) + wave) << 1;
    if (ct0 >= nT) return;                      // wave-uniform
    const bool t1 = (ct0 + 1) < nT;             // second col tile valid
    const bool r1 = (rowBase + 16) < M;         // second row tile valid

    const int ln   = lane & 15;
    const int col0 = (ct0 << 4) + ln;
    const int col1 = t1 ? (col0 + 16) : col0;   // clamp keeps W reads in-bounds
    const int half = (lane >> 4) << 1;          // 0 or 2
    const int rsel = (lane < 16) ? 0 : 8;

    v8f acc00, acc01, acc10, acc11;
    const float bi0 = bias ? bias[col0] : 0.0f;
    const float bi1 = bias ? bias[col1] : 0.0f;
#pragma unroll
    for (int i = 0; i < 8; ++i) { acc00[i] = bi0; acc01[i] = bi1; acc10[i] = bi0; acc11[i] = bi1; }
    if (resid) {
#pragma unroll
        for (int i = 0; i < 8; ++i) {
            size_t r0 = (size_t)(rowBase + rsel + i) * ldc;
            size_t r1r = (size_t)(r1 ? (rowBase + 16 + rsel + i) : (rowBase + rsel + i)) * ldc;
            acc00[i] += resid[r0 + colOff + col0];
            acc01[i] += resid[r0 + colOff + col1];
            acc10[i] += resid[r1r + colOff + col0];
            acc11[i] += resid[r1r + colOff + col1];
        }
    }

    const float* w0 = W + col0;
    const float* w1 = W + col1;
    const int hN  = half * Nc;                  // precomputed row offsets in W
    const int hN1 = hN + Nc;
    const int step = Nc << 2;                   // advance 4 K-rows per iter
    int aoff0 = ln * stride + half;
    int aoff1 = aoff0 + (stride << 4);

#pragma unroll 4
    for (int kk = 0; kk < K; kk += 4) {
        v2f a0 = *(const v2f*)&sA[aoff0];       // 8B aligned
        v2f a1 = *(const v2f*)&sA[aoff1];
        v2f b0, b1;
        b0[0] = w0[hN]; b0[1] = w0[hN1];
        b1[0] = w1[hN]; b1[1] = w1[hN1];
        acc00 = __builtin_amdgcn_wmma_f32_16x16x4_f32(false, a0, false, b0, (short)0, acc00, false, false);
        acc01 = __builtin_amdgcn_wmma_f32_16x16x4_f32(false, a0, false, b1, (short)0, acc01, false, false);
        acc10 = __builtin_amdgcn_wmma_f32_16x16x4_f32(false, a1, false, b0, (short)0, acc10, false, false);
        acc11 = __builtin_amdgcn_wmma_f32_16x16x4_f32(false, a1, false, b1, (short)0, acc11, false, false);
        aoff0 += 4; aoff1 += 4;
        w0 += step; w1 += step;
    }

#pragma unroll
    for (int i = 0; i < 8; ++i) {
        int row0 = rowBase + rsel + i;
        float v00 = acc00[i], v01 = acc01[i];
        if (relu) { v00 = fmaxf(v00, 0.0f); v01 = fmaxf(v01, 0.0f); }
        C[(size_t)row0 * ldc + colOff + col0] = v00;
        if (t1) C[(size_t)row0 * ldc + colOff + col1] = v01;
    }
    if (r1) {
#pragma unroll
        for (int i = 0; i < 8; ++i) {
            int row1 = rowBase + 16 + rsel + i;
            float v10 = acc10[i], v11 = acc11[i];
            if (relu) { v10 = fmaxf(v10, 0.0f); v11 = fmaxf(v11, 0.0f); }
            C[(size_t)row1 * ldc + colOff + col0] = v10;
            if (t1) C[(size_t)row1 * ldc + colOff + col1] = v11;
        }
    }
}

// ----------------------------------------------------------------------
// LayerNorm over last dim 128, one wave (32 lanes x float4) per row.
__global__ __launch_bounds__(256) void layernorm128(
    const float* __restrict__ X, const float* __restrict__ g,
    const float* __restrict__ b, float* __restrict__ Y, int rows)
{
    int row = blockIdx.x * (blockDim.x >> 5) + (threadIdx.x >> 5);
    if (row >= rows) return;
    int lane = threadIdx.x & 31;
    float4 x = ((const float4*)(X + (size_t)row * 128))[lane];
    float s  = x.x + x.y + x.z + x.w;
    float s2 = x.x*x.x + x.y*x.y + x.z*x.z + x.w*x.w;
#pragma unroll
    for (int off = 16; off > 0; off >>= 1) {
        s  += __shfl_xor(s,  off, 32);
        s2 += __shfl_xor(s2, off, 32);
    }
    float mu  = s * (1.0f / 128.0f);
    float var = s2 * (1.0f / 128.0f) - mu * mu;
    float inv = rsqrtf(var + 1e-5f);
    float4 gv = ((const float4*)g)[lane];
    float4 bv = ((const float4*)b)[lane];
    float4 y;
    y.x = (x.x - mu) * inv * gv.x + bv.x;
    y.y = (x.y - mu) * inv * gv.y + bv.y;
    y.z = (x.z - mu) * inv * gv.z + bv.z;
    y.w = (x.w - mu) * inv * gv.w + bv.w;
    ((float4*)(Y + (size_t)row * 128))[lane] = y;
}

// ----------------------------------------------------------------------
// Edge attention: thread per (edge, head). score -> atomic scatter into wV,z.
__global__ __launch_bounds__(256) void edge_attn(
    const float* __restrict__ Q, const float* __restrict__ Km,
    const float* __restrict__ V, const int* __restrict__ src,
    const int* __restrict__ dst, float* __restrict__ wV,
    float* __restrict__ z, int E)
{
    int t = blockIdx.x * blockDim.x + threadIdx.x;
    int e = t >> 3;
    if (e >= E) return;
    int h = t & 7;
    int s = src[e], d = dst[e];
    const float4* kp = (const float4*)(Km + (size_t)s * 128 + h * 16);
    const float4* qp = (const float4*)(Q  + (size_t)d * 128 + h * 16);
    float dot = 0.0f;
#pragma unroll
    for (int i = 0; i < 4; ++i) {
        float4 kv = kp[i], qv = qp[i];
        dot += kv.x*qv.x + kv.y*qv.y + kv.z*qv.z + kv.w*qv.w;
    }
    float sc = expf(fminf(fmaxf(dot * 0.25f, -5.0f), 5.0f));   // inv_sqrt(16)=0.25
    atomicAdd(&z[(size_t)d * NHD + h], sc);
    const float* vp = V  + (size_t)s * 128 + h * 16;
    float*       wp = wV + (size_t)d * 128 + h * 16;
#pragma unroll
    for (int i = 0; i < 16; ++i) atomicAdd(&wp[i], sc * vp[i]);
}

// attn = wV / (z + 1e-6), in place
__global__ __launch_bounds__(256) void attn_norm(
    float* __restrict__ wV, const float* __restrict__ z, long n)
{
    long i = (long)blockIdx.x * blockDim.x + threadIdx.x;
    if (i >= n) return;
    int nrow = (int)(i >> 7);
    int h = (int)((i & 127) >> 4);
    wV[i] = wV[i] / (z[(size_t)nrow * NHD + h] + 1e-6f);
}

// ----------------------------------------------------------------------
// Fused spectral-attention LPE: 2 nodes per 64-thread block, all stages in LDS.
#define NB 2
__global__ __launch_bounds__(64) void lpe_kernel(
    const float* __restrict__ ev, const float* __restrict__ el,
    const float* __restrict__ AW, const float* __restrict__ Ab,
    const float* __restrict__ Wqkv, const float* __restrict__ bqkv,
    const float* __restrict__ Wo, const float* __restrict__ bo,
    const float* __restrict__ g1, const float* __restrict__ b1,
    const float* __restrict__ W1, const float* __restrict__ c1,
    const float* __restrict__ W2, const float* __restrict__ c2,
    const float* __restrict__ g2, const float* __restrict__ b2,
    float* __restrict__ hc)
{
    __shared__ float s_pe [NB][16][17];
    __shared__ float s_qkv[NB][16][49];
    __shared__ float s_ao [NB][16][17];
    __shared__ float s_f1 [NB][16][65];

    const int base = blockIdx.x * NB;
    const int t = threadIdx.x;
    const int T = 64;

    // 1: pe = stack(eigvecs,eigvals) @ A_W + A_b
    for (int i = t; i < NB * 256; i += T) {
        int nd = i >> 8, r = (i >> 4) & 15, c = i & 15;
        int n = base + nd;
        float e1 = (n < NN) ? ev[n * 16 + r] : 0.0f;
        float e2 = (n < NN) ? el[n * 16 + r] : 0.0f;
        s_pe[nd][r][c] = e1 * AW[c] + e2 * AW[16 + c] + Ab[c];
    }
    __syncthreads();
    // 2: qkv = pe @ Wqkv + bqkv   (16 x 48)
    for (int i = t; i < NB * 768; i += T) {
        int nd = i / 768, rem = i - nd * 768, m = rem / 48, j = rem - m * 48;
        float a = bqkv[j];
#pragma unroll
        for (int c = 0; c < 16; ++c) a += s_pe[nd][m][c] * Wqkv[c * 48 + j];
        s_qkv[nd][m][j] = a;
    }
    __syncthreads();
    // 3: 4-head attention, dk=4, softmax over 16 keys
    for (int i = t; i < NB * 64; i += T) {
        int nd = i >> 6, rem = i & 63, mq = rem >> 2, h = rem & 3;
        float q0 = s_qkv[nd][mq][h*4+0], q1 = s_qkv[nd][mq][h*4+1];
        float q2 = s_qkv[nd][mq][h*4+2], q3 = s_qkv[nd][mq][h*4+3];
        float scr[16]; float mx = -1e30f;
#pragma unroll
        for (int mk = 0; mk < 16; ++mk) {
            float dd = q0*s_qkv[nd][mk][16+h*4+0] + q1*s_qkv[nd][mk][16+h*4+1]
                     + q2*s_qkv[nd][mk][16+h*4+2] + q3*s_qkv[nd][mk][16+h*4+3];
            dd *= 0.5f;                       // 1/sqrt(dk=4)
            scr[mk] = dd; mx = fmaxf(mx, dd);
        }
        float ssum = 0.0f;
#pragma unroll
        for (int mk = 0; mk < 16; ++mk) { float p = expf(scr[mk]-mx); scr[mk]=p; ssum+=p; }
        float inv = 1.0f / ssum;
        float o0=0,o1=0,o2=0,o3=0;
#pragma unroll
        for (int mk = 0; mk < 16; ++mk) {
            float p = scr[mk] * inv;
            o0 += p * s_qkv[nd][mk][32+h*4+0];
            o1 += p * s_qkv[nd][mk][32+h*4+1];
            o2 += p * s_qkv[nd][mk][32+h*4+2];
            o3 += p * s_qkv[nd][mk][32+h*4+3];
        }
        s_ao[nd][mq][h*4+0]=o0; s_ao[nd][mq][h*4+1]=o1;
        s_ao[nd][mq][h*4+2]=o2; s_ao[nd][mq][h*4+3]=o3;
    }
    __syncthreads();
    // 4: t = pe + ao @ Wo + bo   -> s_qkv
    for (int i = t; i < NB * 256; i += T) {
        int nd = i >> 8, m = (i >> 4) & 15, c = i & 15;
        float a = s_pe[nd][m][c] + bo[c];
#pragma unroll
        for (int k = 0; k < 16; ++k) a += s_ao[nd][m][k] * Wo[k * 16 + c];
        s_qkv[nd][m][c] = a;
    }
    __syncthreads();
    // 5: LN1 -> s_pe
    for (int i = t; i < NB * 32; i += T) {
        int nd = i >> 5, r = (i >> 1) & 15, hf = i & 1;
        float mu = 0, s2 = 0;
#pragma unroll
        for (int c = 0; c < 16; ++c) { float v = s_qkv[nd][r][c]; mu += v; s2 += v*v; }
        mu *= 0.0625f;
        float var = s2 * 0.0625f - mu * mu;
        float inv = rsqrtf(var + 1e-5f);
        for (int c = hf * 8; c < hf * 8 + 8; ++c)
            s_pe[nd][r][c] = (s_qkv[nd][r][c] - mu) * inv * g1[c] + b1[c];
    }
    __syncthreads();
    // 6: f1 = relu(pe @ W1 + c1)   (16 x 64)
    for (int i = t; i < NB * 1024; i += T) {
        int nd = i >> 10, rem = i & 1023, m = rem >> 6, j = rem & 63;
        float a = c1[j];
#pragma unroll
        for (int c = 0; c < 16; ++c) a += s_pe[nd][m][c] * W1[c * 64 + j];
        s_f1[nd][m][j] = fmaxf(a, 0.0f);
    }
    __syncthreads();
    // 7: u = pe + f1 @ W2 + c2 -> s_qkv
    for (int i = t; i < NB * 256; i += T) {
        int nd = i >> 8, m = (i >> 4) & 15, c = i & 15;
        float a = s_pe[nd][m][c] + c2[c];
#pragma unroll
        for (int j = 0; j < 64; ++j) a += s_f1[nd][m][j] * W2[j * 16 + c];
        s_qkv[nd][m][c] = a;
    }
    __syncthreads();
    // 8: LN2 -> s_pe
    for (int i = t; i < NB * 32; i += T) {
        int nd = i >> 5, r = (i >> 1) & 15, hf = i & 1;
        float mu = 0, s2 = 0;
#pragma unroll
        for (int c = 0; c < 16; ++c) { float v = s_qkv[nd][r][c]; mu += v; s2 += v*v; }
        mu *= 0.0625f;
        float var = s2 * 0.0625f - mu * mu;
        float inv = rsqrtf(var + 1e-5f);
        for (int c = hf * 8; c < hf * 8 + 8; ++c)
            s_pe[nd][r][c] = (s_qkv[nd][r][c] - mu) * inv * g2[c] + b2[c];
    }
    __syncthreads();
    // 9: sum over M -> hc[:, 112:128]
    for (int i = t; i < NB * 16; i += T) {
        int nd = i >> 4, c = i & 15;
        int n = base + nd;
        if (n >= NN) continue;
        float a = 0;
#pragma unroll
        for (int m = 0; m < 16; ++m) a += s_pe[nd][m][c];
        hc[(size_t)n * 128 + 112 + c] = a;
    }
}

// ----------------------------------------------------------------------
// graph mean readout accumulation
__global__ __launch_bounds__(256) void readout_accum(
    const float* __restrict__ hc, const int* __restrict__ gid,
    float* __restrict__ hg, float* __restrict__ cnt, long n)
{
    long i = (long)blockIdx.x * blockDim.x + threadIdx.x;
    if (i >= n) return;
    int node = (int)(i >> 7), c = (int)(i & 127);
    int g = gid[node];
    atomicAdd(&hg[(size_t)g * 128 + c], hc[i]);
    if (c == 0) atomicAdd(&cnt[g], 1.0f);
}

// fused MLP readout: one block per graph
__global__ __launch_bounds__(128) void mlp_readout(
    const float* __restrict__ hg, const float* __restrict__ cnt,
    const float* __restrict__ W0, const float* __restrict__ b0,
    const float* __restrict__ W1, const float* __restrict__ b1,
    const float* __restrict__ W2, const float* __restrict__ b2,
    float* __restrict__ out)
{
    __shared__ float sx[128];
    __shared__ float s0[64];
    __shared__ float s1[32];
    int g = blockIdx.x, t = threadIdx.x;
    float denom = fmaxf(cnt[g], 1.0f);
    sx[t] = hg[(size_t)g * 128 + t] / denom;
    __syncthreads();
    if (t < 64) {
        float a = b0[t];
        for (int c = 0; c < 128; ++c) a += sx[c] * W0[c * 64 + t];
        s0[t] = fmaxf(a, 0.0f);
    }
    __syncthreads();
    if (t < 32) {
        float a = b1[t];
        for (int c = 0; c < 64; ++c) a += s0[c] * W1[c * 32 + t];
        s1[t] = fmaxf(a, 0.0f);
    }
    __syncthreads();
    if (t < NCLS) {
        float a = b2[t];
        for (int c = 0; c < 32; ++c) a += s1[c] * W2[c * NCLS + t];
        out[g * NCLS + t] = a;
    }
}

// ----------------------------------------------------------------------
extern "C" void kernel_launch(void* const* d_in, const int* in_sizes, int n_in,
                              void* d_out, int out_size, void* d_ws, size_t ws_size,
                              hipStream_t stream)
{
    (void)in_sizes; (void)n_in; (void)out_size; (void)ws_size;
    const float* h       = (const float*)d_in[0];
    const float* eigvecs = (const float*)d_in[1];
    const float* eigvals = (const float*)d_in[2];
    const int*   src     = (const int*)d_in[3];
    const int*   dst     = (const int*)d_in[4];
    const int*   gid     = (const int*)d_in[5];
    const float* Wh      = (const float*)d_in[6];
    const float* bh      = (const float*)d_in[7];
    const float* A_W     = (const float*)d_in[8];
    const float* A_b     = (const float*)d_in[9];
    const float* lWqkv   = (const float*)d_in[10];
    const float* lbqkv   = (const float*)d_in[11];
    const float* lWo     = (const float*)d_in[12];
    const float* lbo     = (const float*)d_in[13];
    const float* lg1     = (const float*)d_in[14];
    const float* lb1     = (const float*)d_in[15];
    const float* lW1     = (const float*)d_in[16];
    const float* lc1     = (const float*)d_in[17];
    const float* lW2     = (const float*)d_in[18];
    const float* lc2     = (const float*)d_in[19];
    const float* lg2     = (const float*)d_in[20];
    const float* lb2     = (const float*)d_in[21];
    const float* gWq     = (const float*)d_in[22];
    const float* gbq     = (const float*)d_in[23];
    const float* gWk     = (const float*)d_in[24];
    const float* gbk     = (const float*)d_in[25];
    const float* gWv     = (const float*)d_in[26];
    const float* gbv     = (const float*)d_in[27];
    const float* gWo     = (const float*)d_in[28];
    const float* gbo     = (const float*)d_in[29];
    const float* gln1g   = (const float*)d_in[30];
    const float* gln1b   = (const float*)d_in[31];
    const float* gW1     = (const float*)d_in[32];
    const float* gb1     = (const float*)d_in[33];
    const float* gW2     = (const float*)d_in[34];
    const float* gb2     = (const float*)d_in[35];
    const float* gln2g   = (const float*)d_in[36];
    const float* gln2b   = (const float*)d_in[37];
    const float* mW0     = (const float*)d_in[38];
    const float* mb0     = (const float*)d_in[39];
    const float* mW1     = (const float*)d_in[40];
    const float* mb1     = (const float*)d_in[41];
    const float* mW2     = (const float*)d_in[42];
    const float* mb2     = (const float*)d_in[43];
    float* out = (float*)d_out;

    // workspace carving (floats)
    float* ws   = (float*)d_ws;
    const size_t NP = (size_t)NN * 128;
    float* hc   = ws;
    float* bufQ = hc   + NP;
    float* bufK = bufQ + NP;
    float* bufV = bufK + NP;        // bufK..bufV contiguous -> FFN temp [NN,256]
    float* wV   = bufV + NP;
    float* zbuf = wV   + NP;        // [NN, 8], contiguous after wV
    float* hg   = zbuf + (size_t)NN * NHD;
    float* cnt  = hg   + (size_t)GG * 128;
    float* bufT = bufK;

    const int rowBlocks = (NN + 31) / 32;   // 1563 (last block has 16 rows)
    const dim3 blkG(128);

    // --- LPE positional encoding -> hc[:,112:128] ---
    lpe_kernel<<<NN / NB, 64, 0, stream>>>(eigvecs, eigvals, A_W, A_b,
        lWqkv, lbqkv, lWo, lbo, lg1, lb1, lW1, lc1, lW2, lc2, lg2, lb2, hc);

    // --- node embedding h0 -> hc[:,0:112] ---
    gemm_wmma_f32<<<dim3(rowBlocks, 1), blkG, 0, stream>>>(
        h, Wh, bh, nullptr, hc, NN, 128, 112, 128, 0, 0);

    // --- graph transformer layers ---
    for (int l = 0; l < NL; ++l) {
        const float* Wq = gWq + (size_t)l * 128 * 128;
        const float* Wk = gWk + (size_t)l * 128 * 128;
        const float* Wv = gWv + (size_t)l * 128 * 128;
        const float* Wo = gWo + (size_t)l * 128 * 128;
        const float* W1 = gW1 + (size_t)l * 128 * 256;
        const float* W2 = gW2 + (size_t)l * 256 * 128;
        const float* bq = gbq + l * 128;
        const float* bk = gbk + l * 128;
        const float* bv = gbv + l * 128;
        const float* bo = gbo + l * 128;
        const float* c1 = gb1 + l * 256;
        const float* c2 = gb2 + l * 128;
        const float* n1g = gln1g + l * 128; const float* n1b = gln1b + l * 128;
        const float* n2g = gln2g + l * 128; const float* n2b = gln2b + l * 128;

        gemm_wmma_f32<<<dim3(rowBlocks, 1), blkG, 0, stream>>>(hc, Wq, bq, nullptr, bufQ, NN, 128, 128, 128, 0, 0);
        gemm_wmma_f32<<<dim3(rowBlocks, 1), blkG, 0, stream>>>(hc, Wk, bk, nullptr, bufK, NN, 128, 128, 128, 0, 0);
        gemm_wmma_f32<<<dim3(rowBlocks, 1), blkG, 0, stream>>>(hc, Wv, bv, nullptr, bufV, NN, 128, 128, 128, 0, 0);

        long zn = NP + (long)NN * NHD;   // wV + z (adjacent)
        zero_f32<<<(int)((zn + 255) / 256), 256, 0, stream>>>(wV, zn);

        edge_attn<<<(EE * NHD + 255) / 256, 256, 0, stream>>>(bufQ, bufK, bufV, src, dst, wV, zbuf, EE);
        attn_norm<<<(int)((NP + 255) / 256), 256, 0, stream>>>(wV, zbuf, (long)NP);

        // O-proj + residual, then LN1
        gemm_wmma_f32<<<dim3(rowBlocks, 1), blkG, 0, stream>>>(wV, Wo, bo, hc, bufQ, NN, 128, 128, 128, 0, 0);
        layernorm128<<<(NN + 7) / 8, 256, 0, stream>>>(bufQ, n1g, n1b, hc, NN);

        // FFN: relu(hc@W1+c1) [NN,256] -> (@W2+c2)+hc, LN2
        gemm_wmma_f32<<<dim3(rowBlocks, 2), blkG, 0, stream>>>(hc, W1, c1, nullptr, bufT, NN, 128, 256, 256, 0, 1);
        gemm_wmma_f32<<<dim3(rowBlocks, 1), blkG, 0, stream>>>(bufT, W2, c2, hc, bufQ, NN, 256, 128, 128, 0, 0);
        layernorm128<<<(NN + 7) / 8, 256, 0, stream>>>(bufQ, n2g, n2b, hc, NN);
    }

    // --- mean readout + MLP ---
    long hn = (long)GG * 128 + GG;
    zero_f32<<<(int)((hn + 255) / 256), 256, 0, stream>>>(hg, hn);
    readout_accum<<<(int)((NP + 255) / 256), 256, 0, stream>>>(hc, gid, hg, cnt, (long)NP);
    mlp_readout<<<GG, 128, 0, stream>>>(hg, cnt, mW0, mb0, mW1, mb1, mW2, mb2, out);
}